// RoiPoolingConv_23630910062668
// MI455X (gfx1250) — compile-verified
//
#include <hip/hip_runtime.h>
#include <stdint.h>

// ROI-align 7x7 bilinear pooling, NHWC f32.
// x: (8,100,100,512) f32; rois: (300,4) i32 [x1,y1,w,h]; out: (8,300,7,7,512) f32.
// Memory-bound gather: stage the 4 bilinear source rows (4 x 2KB) per output
// position into a per-wave LDS double buffer via gfx1250 async global->LDS
// loads (ASYNCcnt), overlap with LDS-read + lerp + non-temporal store.

#define POOLSZ 7
#define BDIM   8
#define HDIM   100
#define WDIM   100
#define CDIM   512
#define RDIM   300

typedef float v4f __attribute__((ext_vector_type(4)));

#define WAVES_PER_BLOCK 4
#define THREADS (WAVES_PER_BLOCK * 32)
#define ROWBYTES (CDIM * 4)          /* 2048 B per channel vector */
#define BUFBYTES (4 * ROWBYTES)      /* 8 KB: TL,TR,BL,BR          */
#define LDS_BYTES (WAVES_PER_BLOCK * 2 * BUFBYTES) /* 64 KB */

__device__ __forceinline__ void axis_coords(int p, int start, int size,
                                            int& lo_o, int& hi_o, float& f) {
  float s = (float)size;
  float v = ((float)p + 0.5f) * (s * (1.0f / (float)POOLSZ)) - 0.5f;
  v = fminf(fmaxf(v, 0.0f), s - 1.0f);
  float fl = floorf(v);
  int lo = (int)fl;
  f = v - fl;
  int hi = lo + 1;
  int cap = size - 1;
  if (hi > cap) hi = cap;
  lo_o = start + lo;
  hi_o = start + hi;
}

// Copy one 512-float channel vector (2KB) from global to LDS asynchronously.
// 32 lanes x 16B = 512B per instruction, 4 instructions per row.
__device__ __forceinline__ void async_row(const float* __restrict__ g,
                                          uint32_t lds, int lane) {
  const float* p = g + lane * 4;
  uint32_t l = lds + (uint32_t)lane * 16u;
#pragma unroll
  for (int k = 0; k < 4; ++k) {
    asm volatile("global_load_async_to_lds_b128 %0, %1, off"
                 :: "v"(l), "v"(p)
                 : "memory");
    p += 128;   // 32 lanes * 4 floats
    l += 512;
  }
}

__device__ __forceinline__ void issue_position(const float* rowY0, const float* rowY1,
                                               int X0, int X1, uint32_t lb, int lane) {
  async_row(rowY0 + (size_t)X0 * CDIM, lb + 0 * ROWBYTES, lane);  // TL
  async_row(rowY0 + (size_t)X1 * CDIM, lb + 1 * ROWBYTES, lane);  // TR
  async_row(rowY1 + (size_t)X0 * CDIM, lb + 2 * ROWBYTES, lane);  // BL
  async_row(rowY1 + (size_t)X1 * CDIM, lb + 3 * ROWBYTES, lane);  // BR
}

__global__ __launch_bounds__(THREADS) void roi_align_kernel(
    const float* __restrict__ x, const int* __restrict__ rois,
    float* __restrict__ out) {
  extern __shared__ float smem[];
  const int lane = threadIdx.x & 31;
  const int wave = threadIdx.x >> 5;

  // one wave per (b, r, py) row; 7 px positions pipelined
  const int row_id = blockIdx.x * WAVES_PER_BLOCK + wave;
  const int b  = row_id / (RDIM * POOLSZ);
  const int rr = row_id % (RDIM * POOLSZ);
  const int r  = rr / POOLSZ;
  const int py = rr % POOLSZ;

  const int x1 = rois[r * 4 + 0];
  const int y1 = rois[r * 4 + 1];
  const int w  = rois[r * 4 + 2];
  const int h  = rois[r * 4 + 3];

  int Y0, Y1; float fy;
  axis_coords(py, y1, h, Y0, Y1, fy);
  if (Y0 > HDIM - 1) Y0 = HDIM - 1;
  if (Y1 > HDIM - 1) Y1 = HDIM - 1;

  const float* xb    = x + (size_t)b * HDIM * WDIM * CDIM;
  const float* rowY0 = xb + (size_t)Y0 * WDIM * CDIM;
  const float* rowY1 = xb + (size_t)Y1 * WDIM * CDIM;

  // per-wave LDS double buffer; AS3 flat addr low 32 bits == DS byte offset
  const uint32_t lds0 = (uint32_t)(uintptr_t)(&smem[0]) +
                        (uint32_t)wave * 2u * (uint32_t)BUFBYTES;
  float* sbase = smem + (size_t)wave * 2 * (BUFBYTES / 4);

  // prologue: issue px = 0 into buffer 0
  {
    int X0, X1; float fx0;
    axis_coords(0, x1, w, X0, X1, fx0);
    if (X0 > WDIM - 1) X0 = WDIM - 1;
    if (X1 > WDIM - 1) X1 = WDIM - 1;
    issue_position(rowY0, rowY1, X0, X1, lds0, lane);
  }

  float* outp = out + ((((size_t)b * RDIM + r) * POOLSZ + py) * POOLSZ) * CDIM;

  for (int px = 0; px < POOLSZ; ++px) {
    if (px + 1 < POOLSZ) {
      // prefetch next position into the other buffer
      int X0, X1; float fxn;
      axis_coords(px + 1, x1, w, X0, X1, fxn);
      if (X0 > WDIM - 1) X0 = WDIM - 1;
      if (X1 > WDIM - 1) X1 = WDIM - 1;
      uint32_t lb = lds0 + (uint32_t)((px + 1) & 1) * (uint32_t)BUFBYTES;
      issue_position(rowY0, rowY1, X0, X1, lb, lane);
      // async loads complete in order: allowing 16 outstanding (the batch just
      // issued) guarantees the current position's 16 transfers are done.
      asm volatile("s_wait_asynccnt 16" ::: "memory");
    } else {
      asm volatile("s_wait_asynccnt 0" ::: "memory");
    }

    int dX0, dX1; float fx;                 // only fx needed here
    axis_coords(px, x1, w, dX0, dX1, fx);

    const float* buf = sbase + (size_t)(px & 1) * (BUFBYTES / 4);
    const float* bTL = buf;
    const float* bTR = buf + CDIM;
    const float* bBL = buf + 2 * CDIM;
    const float* bBR = buf + 3 * CDIM;
    float* op = outp + (size_t)px * CDIM;

#pragma unroll
    for (int k = 0; k < 4; ++k) {
      int c = k * 128 + lane * 4;           // 16 floats per lane total
      v4f tl = *(const v4f*)(bTL + c);
      v4f tr = *(const v4f*)(bTR + c);
      v4f bl = *(const v4f*)(bBL + c);
      v4f br = *(const v4f*)(bBR + c);
      v4f top = tl + (tr - tl) * fx;
      v4f bot = bl + (br - bl) * fx;
      v4f res = top + (bot - top) * fy;
      // output is streamed once (241MB): NT store keeps the input set in L2
      __builtin_nontemporal_store(res, (v4f*)(op + c));
    }
  }
}

extern "C" void kernel_launch(void* const* d_in, const int* in_sizes, int n_in,
                              void* d_out, int out_size, void* d_ws, size_t ws_size,
                              hipStream_t stream) {
  (void)in_sizes; (void)n_in; (void)d_ws; (void)ws_size; (void)out_size;
  const float* x   = (const float*)d_in[0];
  const int* rois  = (const int*)d_in[1];
  float* out       = (float*)d_out;

  const int rows   = BDIM * RDIM * POOLSZ;        // 16800 waves
  const int blocks = rows / WAVES_PER_BLOCK;      // 4200 blocks, 128 thr each
  roi_align_kernel<<<blocks, THREADS, LDS_BYTES, stream>>>(x, rois, out);
}